// LocalSelectiveSSMLayer_54992761258347
// MI455X (gfx1250) — compile-verified
//
#include <hip/hip_runtime.h>
#include <math.h>

typedef __attribute__((ext_vector_type(2))) float v2f;
typedef __attribute__((ext_vector_type(8))) float v8f;
typedef __attribute__((ext_vector_type(4))) int   v4i;

#define BSZ   4
#define SEQ   2048
#define DM    1024
#define NST   16
#define MROWS (BSZ*SEQ)       // 8192
#define NTOT  (DM + 2*NST)    // 1056

// ---- GEMM tiling ----
#define BM   128
#define BN   64
#define KC   32
#define NKC  (DM/KC)          // 32 chunks
#define ASTR 36               // padded A row stride (floats): 144B rows, conflict-free
#define BNP  80               // padded B pair-row stride: 160 floats = +32 banks per k-pair

#define NCH  16               // scan chunks
#define TCH  (SEQ/NCH)        // 128 steps per chunk

// ---------------- CDNA5 async global->LDS support ----------------
#if __has_builtin(__builtin_amdgcn_global_load_async_to_lds_b128)
#define HAVE_ASYNC_LDS 1
#else
#define HAVE_ASYNC_LDS 0
#warning "gfx1250 async-to-LDS builtin NOT available; using synchronous fallback"
#endif

typedef __attribute__((address_space(1))) v4i* gv4i_p;   // global
typedef __attribute__((address_space(3))) v4i* lv4i_p;   // LDS

__device__ __forceinline__ void async_copy16(const float* g, float* l) {
#if HAVE_ASYNC_LDS
    __builtin_amdgcn_global_load_async_to_lds_b128((gv4i_p)g, (lv4i_p)l, 0, 0);
#else
    *(float4*)l = *(const float4*)g;
#endif
}

__device__ __forceinline__ void wait_async_zero() {
#if HAVE_ASYNC_LDS
#if __has_builtin(__builtin_amdgcn_s_wait_asynccnt)
    __builtin_amdgcn_s_wait_asynccnt(0);
#else
    asm volatile("s_wait_asynccnt 0x0" ::: "memory");
#endif
#endif
}

// =====================================================================
// Kernel 1: fused projection GEMM  [delta_pre | Bm | Cm] = x @ [Wd|Wb|Wc]
// f32 WMMA 16x16x4. 128x64 block tile, 8 waves (wave: 16M x 64N).
// A tile: async global->LDS (ASYNCcnt). B tile: reg prefetch, stored
// k-paired in LDS so each WMMA B-fragment is one aligned b64 DS load.
// =====================================================================
__global__ __launch_bounds__(256)
void proj_gemm_kernel(const float* __restrict__ x,
                      const float* __restrict__ Wd, const float* __restrict__ bd,
                      const float* __restrict__ Wb, const float* __restrict__ bb,
                      const float* __restrict__ Wc, const float* __restrict__ bc,
                      float* __restrict__ delta, float* __restrict__ Bm,
                      float* __restrict__ Cm)
{
    __shared__ float As[2][BM][ASTR];           // 36,864 B
    __shared__ float BsP[2][KC/2][BNP][2];      // 20,480 B  (k-paired)

    const int tid  = threadIdx.x;
    const int wave = tid >> 5;
    const int lane = tid & 31;
    const int m0blk = blockIdx.y * BM;
    const int j0blk = blockIdx.x * BN;

    v8f acc0 = {}, acc1 = {}, acc2 = {}, acc3 = {};
    float4 bPre[2];

    // ---- A tile: 128 rows x 32 cols, 256 thr * 4 x b128 async copies ----
    auto issueA = [&](int buf, int k0) {
        const int c4 = tid & 7;      // 8 x float4 per row
        const int r  = tid >> 3;     // 32 rows per pass
        #pragma unroll
        for (int p = 0; p < 4; ++p) {
            const int row = r + p * 32;
            async_copy16(x + (size_t)(m0blk + row) * DM + k0 + c4 * 4,
                         &As[buf][row][c4 * 4]);
        }
    };

    // ---- B tile: 32 rows x 64 cols of [Wd|Wb|Wc], 256 thr * 2 x float4 ----
    auto loadBregs = [&](int k0) {
        const int bc4 = tid & 15;    // 16 float4 per row
        const int br  = tid >> 4;    // 16 rows per pass
        #pragma unroll
        for (int p = 0; p < 2; ++p) {
            const int k = k0 + br + p * 16;
            const int j = j0blk + bc4 * 4;
            float4 v = make_float4(0.f, 0.f, 0.f, 0.f);
            if (j < DM) {
                v = *reinterpret_cast<const float4*>(Wd + (size_t)k * DM + j);
            } else if (j < DM + NST) {
                v = *reinterpret_cast<const float4*>(Wb + (size_t)k * NST + (j - DM));
            } else if (j < NTOT) {
                v = *reinterpret_cast<const float4*>(Wc + (size_t)k * NST + (j - DM - NST));
            }
            bPre[p] = v;
        }
    };

    auto commitB = [&](int buf) {
        const int bc4 = tid & 15, br = tid >> 4;
        #pragma unroll
        for (int p = 0; p < 2; ++p) {
            const int kl = br + p * 16;
            const int kk = kl >> 1, par = kl & 1;
            float bv[4];
            *(float4*)bv = bPre[p];
            #pragma unroll
            for (int i = 0; i < 4; ++i)
                BsP[buf][kk][bc4 * 4 + i][par] = bv[i];
        }
    };

    const int n    = lane & 15;          // A row / B col within 16x16 tile
    const int kbh  = lane >> 4;          // K half-select per ISA fragment layout
    const int mrow = wave * 16 + n;

    issueA(0, 0);
    loadBregs(0);

    int buf = 0;
    for (int kc = 0; kc < NKC; ++kc) {
        commitB(buf);
        wait_async_zero();               // this wave's A-tile asyncs landed
        __syncthreads();                 // whole block: LDS chunk kc ready
        if (kc + 1 < NKC) {
            issueA(buf ^ 1, (kc + 1) * KC);   // safe: all waves past barrier
            loadBregs((kc + 1) * KC);
        }

        #pragma unroll
        for (int ks = 0; ks < KC; ks += 4) {
            const int kk = (ks >> 1) + kbh;                 // k-pair index
            v2f a = *(const v2f*)&As[buf][mrow][ks + 2 * kbh];
            v2f b;
            b = *(const v2f*)&BsP[buf][kk][0 * 16 + n][0];
            acc0 = __builtin_amdgcn_wmma_f32_16x16x4_f32(false, a, false, b,
                                                         (short)0, acc0, false, false);
            b = *(const v2f*)&BsP[buf][kk][1 * 16 + n][0];
            acc1 = __builtin_amdgcn_wmma_f32_16x16x4_f32(false, a, false, b,
                                                         (short)0, acc1, false, false);
            b = *(const v2f*)&BsP[buf][kk][2 * 16 + n][0];
            acc2 = __builtin_amdgcn_wmma_f32_16x16x4_f32(false, a, false, b,
                                                         (short)0, acc2, false, false);
            b = *(const v2f*)&BsP[buf][kk][3 * 16 + n][0];
            acc3 = __builtin_amdgcn_wmma_f32_16x16x4_f32(false, a, false, b,
                                                         (short)0, acc3, false, false);
        }
        buf ^= 1;
    }

    // ---- epilogue: bias (+softplus for delta), scatter per 16-col segment ----
    const int rbase = m0blk + wave * 16 + (lane >> 4) * 8;  // C/D layout: 8 rows/lane
    v8f accs[4] = {acc0, acc1, acc2, acc3};
    #pragma unroll
    for (int t = 0; t < 4; ++t) {
        const int jb = j0blk + t * 16;
        if (jb >= NTOT) break;
        const int j = jb + n;
        if (jb < DM) {
            const float bias = bd[j];
            #pragma unroll
            for (int v = 0; v < 8; ++v) {
                float val = accs[t][v] + bias;
                val = (val > 20.f) ? val : log1pf(expf(val));   // softplus
                delta[(size_t)(rbase + v) * DM + j] = val;
            }
        } else if (jb < DM + NST) {
            const float bias = bb[j - DM];
            #pragma unroll
            for (int v = 0; v < 8; ++v)
                Bm[(size_t)(rbase + v) * NST + (j - DM)] = accs[t][v] + bias;
        } else {
            const float bias = bc[j - DM - NST];
            #pragma unroll
            for (int v = 0; v < 8; ++v)
                Cm[(size_t)(rbase + v) * NST + (j - DM - NST)] = accs[t][v] + bias;
        }
    }
}

// =====================================================================
// Chunked linear-recurrence scan: h_t = exp(dt*A)*h + dt*x*B
// Pass 1: per (b,d,chunk): local end state hL and decay product P
// =====================================================================
__global__ __launch_bounds__(256)
void scan_pass1(const float* __restrict__ x, const float* __restrict__ delta,
                const float* __restrict__ Bm, const float* __restrict__ A_log,
                float* __restrict__ Pend, float* __restrict__ hLend)
{
    const int gid = blockIdx.x * blockDim.x + threadIdx.x; // B*NCH*DM threads
    const int d = gid & (DM - 1);
    const int c = (gid >> 10) & (NCH - 1);
    const int b = gid >> 14;

    float A[NST], h[NST], P[NST];
    #pragma unroll
    for (int nn = 0; nn < NST; ++nn) {
        A[nn] = -__expf(A_log[d * NST + nn]);
        h[nn] = 0.f;
        P[nn] = 1.f;
    }

    const int t0 = c * TCH;
    const float* dptr = delta + ((size_t)b * SEQ + t0) * DM + d;
    const float* xptr = x     + ((size_t)b * SEQ + t0) * DM + d;
    const float* bptr = Bm    + ((size_t)b * SEQ + t0) * NST;

    for (int s = 0; s < TCH; ++s) {
        const float dt = dptr[(size_t)s * DM];
        const float xv = xptr[(size_t)s * DM];
        float Bv[NST];
        *(float4*)&Bv[0]  = *(const float4*)(bptr + (size_t)s * NST + 0);
        *(float4*)&Bv[4]  = *(const float4*)(bptr + (size_t)s * NST + 4);
        *(float4*)&Bv[8]  = *(const float4*)(bptr + (size_t)s * NST + 8);
        *(float4*)&Bv[12] = *(const float4*)(bptr + (size_t)s * NST + 12);
        const float du = dt * xv;
        #pragma unroll
        for (int nn = 0; nn < NST; ++nn) {
            const float ba = __expf(dt * A[nn]);
            h[nn] = ba * h[nn] + du * Bv[nn];
            P[nn] *= ba;
        }
    }
    const size_t o = (((size_t)b * NCH + c) * DM + d) * NST;
    #pragma unroll
    for (int nn = 0; nn < NST; ++nn) {
        Pend[o + nn]  = P[nn];
        hLend[o + nn] = h[nn];
    }
}

// Pass 2: serial stitch across the 16 chunks per (b,d,n)
__global__ __launch_bounds__(256)
void scan_pass2(const float* __restrict__ Pend, const float* __restrict__ hLend,
                float* __restrict__ Hstart)
{
    const int gid = blockIdx.x * blockDim.x + threadIdx.x; // B*DM*NST threads
    const int nn = gid & (NST - 1);
    const int d  = (gid >> 4) & (DM - 1);
    const int b  = gid >> 14;
    float H = 0.f;
    #pragma unroll
    for (int c = 0; c < NCH; ++c) {
        const size_t idx = (((size_t)b * NCH + c) * DM + d) * NST + nn;
        Hstart[idx] = H;
        H = hLend[idx] + Pend[idx] * H;
    }
}

// Pass 3: re-run recurrence per chunk from corrected start state, emit y
__global__ __launch_bounds__(256)
void scan_pass3(const float* __restrict__ x, const float* __restrict__ delta,
                const float* __restrict__ Bm, const float* __restrict__ Cm,
                const float* __restrict__ A_log, const float* __restrict__ D_skip,
                const float* __restrict__ Hstart, float* __restrict__ y)
{
    const int gid = blockIdx.x * blockDim.x + threadIdx.x;
    const int d = gid & (DM - 1);
    const int c = (gid >> 10) & (NCH - 1);
    const int b = gid >> 14;

    float A[NST], h[NST];
    const size_t hs = (((size_t)b * NCH + c) * DM + d) * NST;
    #pragma unroll
    for (int nn = 0; nn < NST; ++nn) {
        A[nn] = -__expf(A_log[d * NST + nn]);
        h[nn] = Hstart[hs + nn];
    }
    const float Dsk = D_skip[d];

    const int t0 = c * TCH;
    const float* dptr = delta + ((size_t)b * SEQ + t0) * DM + d;
    const float* xptr = x     + ((size_t)b * SEQ + t0) * DM + d;
    const float* bptr = Bm    + ((size_t)b * SEQ + t0) * NST;
    const float* cptr = Cm    + ((size_t)b * SEQ + t0) * NST;
    float* yptr       = y     + ((size_t)b * SEQ + t0) * DM + d;

    for (int s = 0; s < TCH; ++s) {
        const float dt = dptr[(size_t)s * DM];
        const float xv = xptr[(size_t)s * DM];
        float Bv[NST], Cv[NST];
        *(float4*)&Bv[0]  = *(const float4*)(bptr + (size_t)s * NST + 0);
        *(float4*)&Bv[4]  = *(const float4*)(bptr + (size_t)s * NST + 4);
        *(float4*)&Bv[8]  = *(const float4*)(bptr + (size_t)s * NST + 8);
        *(float4*)&Bv[12] = *(const float4*)(bptr + (size_t)s * NST + 12);
        *(float4*)&Cv[0]  = *(const float4*)(cptr + (size_t)s * NST + 0);
        *(float4*)&Cv[4]  = *(const float4*)(cptr + (size_t)s * NST + 4);
        *(float4*)&Cv[8]  = *(const float4*)(cptr + (size_t)s * NST + 8);
        *(float4*)&Cv[12] = *(const float4*)(cptr + (size_t)s * NST + 12);
        const float du = dt * xv;
        float yv = 0.f;
        #pragma unroll
        for (int nn = 0; nn < NST; ++nn) {
            const float ba = __expf(dt * A[nn]);
            h[nn] = ba * h[nn] + du * Bv[nn];
            yv += h[nn] * Cv[nn];
        }
        yv += xv * Dsk;
        yptr[(size_t)s * DM] = yv;
    }
}

// =====================================================================
extern "C" void kernel_launch(void* const* d_in, const int* in_sizes, int n_in,
                              void* d_out, int out_size, void* d_ws, size_t ws_size,
                              hipStream_t stream)
{
    const float* x      = (const float*)d_in[0];
    const float* A_log  = (const float*)d_in[1];
    const float* D_skip = (const float*)d_in[2];
    const float* Wd     = (const float*)d_in[3];
    const float* bd     = (const float*)d_in[4];
    const float* Wb     = (const float*)d_in[5];
    const float* bb     = (const float*)d_in[6];
    const float* Wc     = (const float*)d_in[7];
    const float* bc     = (const float*)d_in[8];
    float* y = (float*)d_out;

    float* ws = (float*)d_ws;
    float* delta = ws;                                   // 8192*1024
    float* Bm    = delta + (size_t)MROWS * DM;           // 8192*16
    float* Cm    = Bm + (size_t)MROWS * NST;             // 8192*16
    float* Pend  = Cm + (size_t)MROWS * NST;             // 4*16*1024*16
    float* hLend = Pend + (size_t)BSZ * NCH * DM * NST;
    float* Hstart= hLend + (size_t)BSZ * NCH * DM * NST;

    dim3 gGemm((NTOT + BN - 1) / BN, MROWS / BM);        // (17, 64)
    proj_gemm_kernel<<<gGemm, 256, 0, stream>>>(x, Wd, bd, Wb, bb, Wc, bc,
                                                delta, Bm, Cm);

    const int scanThreads = BSZ * NCH * DM;              // 65536
    scan_pass1<<<scanThreads / 256, 256, 0, stream>>>(x, delta, Bm, A_log,
                                                      Pend, hLend);
    scan_pass2<<<(BSZ * DM * NST) / 256, 256, 0, stream>>>(Pend, hLend, Hstart);
    scan_pass3<<<scanThreads / 256, 256, 0, stream>>>(x, delta, Bm, Cm, A_log,
                                                      D_skip, Hstart, y);
}